// TemporalSmoother_32418413150863
// MI455X (gfx1250) — compile-verified
//
#include <hip/hip_runtime.h>
#include <hip/hip_bf16.h>

typedef __attribute__((ext_vector_type(2))) float v2f;
typedef __attribute__((ext_vector_type(8))) float v8f;

#define ALPHA 0.85f
#define BETA  0.15f
#define TLEN  512
#define DLEN  256

// One wave = one (b,c) sequence x 16-wide d-tile. Chunked EMA scan:
//   chunk Z = L(16x16, lower-tri Toeplitz) * X(16t x 16d) via 4x v_wmma_f32_16x16x4_f32,
//   then Y = Z + alpha^{k+1} (outer) y_prev, carry via ds_bpermute of D-row 15.
__global__ __launch_bounds__(256) void ema_scan_wmma_kernel(
    const float* __restrict__ x, float* __restrict__ y, int tilesPerBC) {
  const int lane = threadIdx.x & 31;
  const int wave = threadIdx.x >> 5;
  const int tile = blockIdx.x * 8 + wave;      // global 16-wide d-tile id
  const int bc   = tile / tilesPerBC;          // which (b,c) sequence
  const int dt   = tile % tilesPerBC;          // which d-tile
  const int col  = lane & 15;                  // N (d within tile) / M for A
  const int half = lane >> 4;                  // lane group 0/1

  const int base = bc * (TLEN * DLEN) + dt * 16;     // float index of (t=0, d0)
  const float* __restrict__ px = x + base + col + half * 2 * DLEN;  // B-operand lanes
  float* __restrict__ py       = y + base + col + half * 8 * DLEN;  // D-matrix lanes

  // --- A operand = L matrix, per ISA 16x4 f32 layout: VGPR j, lane L holds
  //     K = j + 2*(L/16) (local), M = L%16.  Global t' = 4k + Klocal.
  v2f A[4];
#pragma unroll
  for (int k = 0; k < 4; ++k) {
#pragma unroll
    for (int j = 0; j < 2; ++j) {
      const int tp = 4 * k + j + 2 * half;     // t' column of L
      const int e  = col - tp;                 // exponent M - t'
      float p = BETA;                          // branchless beta * alpha^e
#pragma unroll
      for (int i = 0; i < 15; ++i) p = (i < e) ? p * ALPHA : p;
      A[k][j] = (e >= 0) ? p : 0.0f;
    }
  }

  // --- carry factors: D VGPR r is row M = r + 8*half; factor = alpha^{M+1}
  float a8 = ALPHA * ALPHA; a8 *= a8; a8 *= a8;       // alpha^8 (const-folded)
  float carry[8];
  carry[0] = ALPHA * (half ? a8 : 1.0f);
#pragma unroll
  for (int r = 1; r < 8; ++r) carry[r] = carry[r - 1] * ALPHA;

  // --- seed y[-1] = x[0] so the uniform recurrence yields y[0] = x[0] exactly
  float yprev = x[base + col];

#pragma unroll 2
  for (int c0 = 0; c0 < TLEN / 16; ++c0) {
    const int t0off = c0 * 16 * DLEN;

    // Load B operand (X chunk), 4x v2f; layout mirrors A: row t' = 4k + j + 2*half.
    v2f B[4];
#pragma unroll
    for (int k = 0; k < 4; ++k) {
      B[k][0] = __builtin_nontemporal_load(px + t0off + (4 * k + 0) * DLEN);
      B[k][1] = __builtin_nontemporal_load(px + t0off + (4 * k + 1) * DLEN);
    }

    // Z = L * X : 4 chained f32 WMMAs over K = 16
    v8f Cm = {};
    Cm = __builtin_amdgcn_wmma_f32_16x16x4_f32(false, A[0], false, B[0], (short)0, Cm, false, false);
    Cm = __builtin_amdgcn_wmma_f32_16x16x4_f32(false, A[1], false, B[1], (short)0, Cm, false, false);
    Cm = __builtin_amdgcn_wmma_f32_16x16x4_f32(false, A[2], false, B[2], (short)0, Cm, false, false);
    Cm = __builtin_amdgcn_wmma_f32_16x16x4_f32(false, A[3], false, B[3], (short)0, Cm, false, false);

    // Rank-1 carry update + streaming store (D VGPR r -> t = t0 + r + 8*half)
#pragma unroll
    for (int r = 0; r < 8; ++r) {
      Cm[r] += carry[r] * yprev;
      __builtin_nontemporal_store(Cm[r], py + t0off + r * DLEN);
    }

    // Next carry: y[t0+15] (M=15) lives in D VGPR7, lanes 16..31, N = lane-16.
    const int iv = __builtin_amdgcn_ds_bpermute((16 + col) << 2, __float_as_int(Cm[7]));
    yprev = __int_as_float(iv);
  }
}

extern "C" void kernel_launch(void* const* d_in, const int* in_sizes, int n_in,
                              void* d_out, int out_size, void* d_ws, size_t ws_size,
                              hipStream_t stream) {
  (void)n_in; (void)d_ws; (void)ws_size; (void)out_size;
  const float* x = (const float*)d_in[0];
  float* y = (float*)d_out;

  const int total      = in_sizes[0];              // 16*32*512*256 = 67,108,864
  const int tilesPerBC = DLEN / 16;                // 16
  const int nTiles     = total / (TLEN * 16);      // 8192 waves
  const int nBlocks    = nTiles / 8;               // 8 waves (256 thr) per block

  ema_scan_wmma_kernel<<<dim3(nBlocks), dim3(256), 0, stream>>>(x, y, tilesPerBC);
}